// BSRNN_83313775608028
// MI455X (gfx1250) — compile-verified
//
#include <hip/hip_runtime.h>

typedef __attribute__((ext_vector_type(2))) float v2f;
typedef __attribute__((ext_vector_type(8))) float v8f;

#define CH_STRIDE 8196000LL   /* 4098*2000 */
#define NB 10

__device__ __forceinline__ float sigf(float v) { return 1.0f / (1.0f + expf(-v)); }
__device__ __forceinline__ float lrelu(float v) { return v > 0.0f ? v : 0.01f * v; }

#define V8F_ZERO {0.f, 0.f, 0.f, 0.f, 0.f, 0.f, 0.f, 0.f}

// Fully-unrolled 16x16 f32 dot over compile-time K with dual accumulator chains.
// a_base: LDS, already offset to [m][kg] (row stride given by LDA template arg is
//         baked by caller into a_base addressing -> we pass stride-1 k walk).
// b_base: global, already offset to [kg][n], k stride = ldb.
template <int K>
__device__ __forceinline__ v8f wmma_dot(const float* a_base,
                                        const float* __restrict__ b_base, int ldb) {
  v8f acc0 = V8F_ZERO;
  v8f acc1 = V8F_ZERO;
#pragma unroll
  for (int k = 0; k < K; k += 8) {
    v2f a0, b0, a1, b1;
    a0[0] = a_base[k];
    a0[1] = a_base[k + 1];
    b0[0] = b_base[k * ldb];
    b0[1] = b_base[(k + 1) * ldb];
    a1[0] = a_base[k + 4];
    a1[1] = a_base[k + 5];
    b1[0] = b_base[(k + 4) * ldb];
    b1[1] = b_base[(k + 5) * ldb];
    acc0 = __builtin_amdgcn_wmma_f32_16x16x4_f32(false, a0, false, b0, (short)0, acc0,
                                                 false, false);
    acc1 = __builtin_amdgcn_wmma_f32_16x16x4_f32(false, a1, false, b1, (short)0, acc1,
                                                 false, false);
  }
  return acc0 + acc1;
}

// ---------------------------------------------------------------------------
// Kernel 1: mean over channels + per-band 2-layer MLP -> z[(t*10+band)*64]
// grid (125), block 128 (4 waves; wave w owns output cols w*16..w*16+15)
// ---------------------------------------------------------------------------
__global__ void band_in_kernel(const float* __restrict__ x,
                               const float* __restrict__ W1, const float* __restrict__ b1,
                               const float* __restrict__ W2, const float* __restrict__ b2,
                               float* __restrict__ z, int band, int off, int K1) {
  __shared__ float As[16 * 64];
  __shared__ float H1s[16 * 64];
  const int tid = threadIdx.x;
  const int lane = tid & 31;
  const int wave = tid >> 5;
  const int tbase = blockIdx.x * 16;
  const int m = lane & 15, hi = lane >> 4, kg = hi * 2;
  const int nb = wave * 16;

  v8f acc = V8F_ZERO;
  for (int k0 = 0; k0 < K1; k0 += 64) {
    const int kc = (K1 - k0 > 64) ? 64 : (K1 - k0);
    // load/pad A chunk (16 t-rows x 64 k-cols), coalesced along t
    for (int idx = tid; idx < 16 * 64; idx += 128) {
      int kk = idx >> 4, tt = idx & 15;
      float v = 0.0f;
      if (kk < kc) {
        long long fi = (long long)(off + k0 + kk) * 2000 + tbase + tt;
        v = 0.5f * (x[fi] + x[fi + CH_STRIDE]);
      }
      As[tt * 64 + kk] = v;
    }
    __syncthreads();
    if (kc == 64) {
      // fast fully-unrolled path
      v8f part = wmma_dot<64>(As + m * 64 + kg, W1 + (k0 + kg) * 64 + nb + m, 64);
      acc = acc + part;
    } else {
      // tail chunk: A zero-padded, clamp B row index to stay in-bounds
      int kcr = (kc + 3) & ~3;
      const float* a_base = As + m * 64 + kg;
      for (int k = 0; k < kcr; k += 4) {
        v2f a, b;
        a[0] = a_base[k];
        a[1] = a_base[k + 1];
        int kA = k0 + k + kg;
        int kB = kA + 1;
        int kAc = kA < K1 ? kA : K1 - 1;
        int kBc = kB < K1 ? kB : K1 - 1;
        b[0] = W1[kAc * 64 + nb + m];
        b[1] = W1[kBc * 64 + nb + m];
        acc = __builtin_amdgcn_wmma_f32_16x16x4_f32(false, a, false, b, (short)0, acc,
                                                    false, false);
      }
    }
    __syncthreads();
  }
  // H1 = lrelu(acc + b1)
  float bb = b1[nb + m];
  for (int r = 0; r < 8; ++r)
    H1s[(r + 8 * hi) * 64 + nb + m] = lrelu(acc[r] + bb);
  __syncthreads();
  // Z = H1 @ W2 + b2
  v8f acc2 = wmma_dot<64>(H1s + m * 64 + kg, W2 + kg * 64 + nb + m, 64);
  float b2v = b2[nb + m];
  for (int r = 0; r < 8; ++r) {
    long long t = tbase + r + 8 * hi;
    z[(t * NB + band) * 64 + nb + m] = acc2[r] + b2v;
  }
}

// ---------------------------------------------------------------------------
// Generic WMMA GEMM: out[r, n] = act( A[r,:]@B + bias(+bias2) ) (+ res[r,n])
// A row r addressed as A + r*lda + aoff.  K (template) in {64,128,256};
// N multiple of 64; M multiple of 16.
// grid (M/16, N/64), block 128 (4 waves; wave w owns 16 output cols).
// ---------------------------------------------------------------------------
template <int K>
__global__ void gemm_kernel(const float* __restrict__ A, int lda, int aoff,
                            const float* __restrict__ B, int ldb,
                            const float* __restrict__ bias, const float* __restrict__ bias2,
                            const float* __restrict__ res, int ldres,
                            float* __restrict__ out, int ldo, int act) {
  __shared__ float As[16 * K];
  const int tid = threadIdx.x;
  const int lane = tid & 31;
  const int wave = tid >> 5;
  const long long mbase = (long long)blockIdx.x * 16;
  const int nbase = blockIdx.y * 64 + wave * 16;
  for (int idx = tid; idx < 16 * K; idx += 128) {
    int r = idx / K, k = idx - r * K;
    As[r * K + k] = A[(mbase + r) * lda + aoff + k];
  }
  __syncthreads();
  const int m = lane & 15, hi = lane >> 4, kg = hi * 2;
  v8f acc = wmma_dot<K>(As + m * K + kg, B + kg * ldb + nbase + m, ldb);
  float bs = bias ? bias[nbase + m] : 0.0f;
  if (bias2) bs += bias2[nbase + m];
  for (int r = 0; r < 8; ++r) {
    long long row = mbase + r + 8 * hi;
    float v = acc[r] + bs;
    if (act == 1) v = lrelu(v);
    if (res) v += res[row * ldres + nbase + m];
    out[row * ldo + nbase + m] = v;
  }
}

// ---------------------------------------------------------------------------
// Bandwise (bidirectional) LSTM: seq over 10 bands, batch = 2000 t's.
// Each block owns 16 t's.  grid (125), block 256 (8 waves x 2 gate N-tiles).
// in rows: (t*10 + band)*INW ; out rows: (t*10 + band)*outW + oofs.
// ---------------------------------------------------------------------------
template <int INW>
__global__ void lstm_band_kernel(const float* __restrict__ in,
                                 const float* __restrict__ Wih, const float* __restrict__ Whh,
                                 const float* __restrict__ bih, const float* __restrict__ bhh,
                                 float* __restrict__ outp, int outW, int oofs, int rev) {
  __shared__ float xin[16 * INW];
  __shared__ float hs[16 * 64];
  __shared__ float cs[16 * 64];
  __shared__ float gs[16 * 256];
  const int tid = threadIdx.x;
  const int lane = tid & 31;
  const int wave = tid >> 5;
  const int tbase = blockIdx.x * 16;
  const int m = lane & 15, hi = lane >> 4, kg = hi * 2;
  for (int i = tid; i < 16 * 64; i += 256) { hs[i] = 0.0f; cs[i] = 0.0f; }
  __syncthreads();

  for (int step = 0; step < 10; ++step) {
    const int s = rev ? 9 - step : step;
    for (int idx = tid; idx < 16 * INW; idx += 256) {
      int tt = idx / INW, k = idx - tt * INW;
      xin[tt * INW + k] = in[((long long)(tbase + tt) * NB + s) * INW + k];
    }
    __syncthreads();
    for (int half = 0; half < 2; ++half) {
      int nb = (wave * 2 + half) * 16;
      v8f accx = wmma_dot<INW>(xin + m * INW + kg, Wih + kg * 256 + nb + m, 256);
      v8f acch = wmma_dot<64>(hs + m * 64 + kg, Whh + kg * 256 + nb + m, 256);
      float bb = bih[nb + m] + bhh[nb + m];
      for (int r = 0; r < 8; ++r)
        gs[(r + 8 * hi) * 256 + nb + m] = accx[r] + acch[r] + bb;
    }
    __syncthreads();
    for (int idx = tid; idx < 16 * 64; idx += 256) {
      int row = idx >> 6, j = idx & 63;
      float ig = gs[row * 256 + j];
      float fg = gs[row * 256 + 64 + j];
      float gg = gs[row * 256 + 128 + j];
      float og = gs[row * 256 + 192 + j];
      float cv = sigf(fg) * cs[idx] + sigf(ig) * tanhf(gg);
      float hv = sigf(og) * tanhf(cv);
      cs[idx] = cv;
      hs[idx] = hv;
      outp[((long long)(tbase + row) * NB + s) * outW + oofs + j] = hv;
    }
    __syncthreads();
  }
}

// ---------------------------------------------------------------------------
// Timewise (unidirectional) LSTM recurrence: 2000 steps, batch = 10 bands.
// xp = x@Wih + bih + bhh precomputed (rows (t*10+band)*256).
// Single persistent block; rows 10..15 of the 16-row tile are inert zeros.
// ---------------------------------------------------------------------------
__global__ void lstm_time_kernel(const float* __restrict__ xp,
                                 const float* __restrict__ Whh,
                                 float* __restrict__ outp) {
  __shared__ float hs[16 * 64];
  __shared__ float cs[16 * 64];
  __shared__ float gs[16 * 256];
  const int tid = threadIdx.x;
  const int lane = tid & 31;
  const int wave = tid >> 5;
  const int m = lane & 15, hi = lane >> 4, kg = hi * 2;
  for (int i = tid; i < 16 * 64; i += 256) { hs[i] = 0.0f; cs[i] = 0.0f; }
  __syncthreads();

  for (int t = 0; t < 2000; ++t) {
    for (int half = 0; half < 2; ++half) {
      int nb = (wave * 2 + half) * 16;
      v8f acc = wmma_dot<64>(hs + m * 64 + kg, Whh + kg * 256 + nb + m, 256);
      for (int r = 0; r < 8; ++r) {
        int band = r + 8 * hi;
        float xv = (band < NB) ? xp[((long long)t * NB + band) * 256 + nb + m] : 0.0f;
        gs[band * 256 + nb + m] = acc[r] + xv;
      }
    }
    __syncthreads();
    for (int idx = tid; idx < 16 * 64; idx += 256) {
      int row = idx >> 6, j = idx & 63;
      float ig = gs[row * 256 + j];
      float fg = gs[row * 256 + 64 + j];
      float gg = gs[row * 256 + 128 + j];
      float og = gs[row * 256 + 192 + j];
      float cv = sigf(fg) * cs[idx] + sigf(ig) * tanhf(gg);
      float hv = sigf(og) * tanhf(cv);
      cs[idx] = cv;
      hs[idx] = hv;
      if (row < NB) outp[((long long)t * NB + row) * 64 + j] = hv;
    }
    __syncthreads();
  }
}

// ---------------------------------------------------------------------------
// Final per-band mask head: H3 = H2 @ W3 + b3 ; GLU(a, g) ; out = x * mask.
// halfN = 2w (mask width), N = 4w.  grid (125, ceil(halfN/64)), block 128.
// ---------------------------------------------------------------------------
__global__ void maskout_kernel(const float* __restrict__ H2,
                               const float* __restrict__ W3, int N,
                               const float* __restrict__ b3,
                               const float* __restrict__ x,
                               float* __restrict__ out,
                               int off, int halfN) {
  __shared__ float As[16 * 256];
  __shared__ float mt[4][256];
  const int tid = threadIdx.x;
  const int lane = tid & 31;
  const int wave = tid >> 5;
  const int tbase = blockIdx.x * 16;
  const int jbase = blockIdx.y * 64 + wave * 16;
  for (int idx = tid; idx < 16 * 256; idx += 128) {
    int r = idx >> 8, k = idx & 255;
    As[idx] = H2[(long long)(tbase + r) * 256 + k];
  }
  __syncthreads();
  const int m = lane & 15, hi = lane >> 4, kg = hi * 2;
  if (jbase < halfN) {                       // wave-uniform predicate
    int ja = jbase + m; if (ja >= halfN) ja = halfN - 1;   // clamp B column reads
    int jg = ja + halfN;
    const float* ab = As + m * 256 + kg;
    v8f aAcc = V8F_ZERO;
    v8f gAcc = V8F_ZERO;
#pragma unroll 8
    for (int k = 0; k < 256; k += 4) {
      v2f a, b, c;
      a[0] = ab[k];
      a[1] = ab[k + 1];
      b[0] = W3[(k + kg) * N + ja];
      b[1] = W3[(k + kg + 1) * N + ja];
      c[0] = W3[(k + kg) * N + jg];
      c[1] = W3[(k + kg + 1) * N + jg];
      aAcc = __builtin_amdgcn_wmma_f32_16x16x4_f32(false, a, false, b, (short)0, aAcc,
                                                   false, false);
      gAcc = __builtin_amdgcn_wmma_f32_16x16x4_f32(false, a, false, c, (short)0, gAcc,
                                                   false, false);
    }
    float ba = b3[ja], bg = b3[jg];
    for (int r = 0; r < 8; ++r) {
      float av = aAcc[r] + ba;
      float gv = gAcc[r] + bg;
      mt[wave][m * 16 + r + 8 * hi] = av * sigf(gv);   // transposed: [j][t]
    }
    // coalesced store: consecutive lanes -> consecutive t
    for (int idx = lane; idx < 256; idx += 32) {
      int jj = idx >> 4, tt = idx & 15;
      int j = jbase + jj;
      if (j < halfN) {
        long long fi = (long long)(off + j) * 2000 + tbase + tt;
        float mk = mt[wave][jj * 16 + tt];
        out[fi] = x[fi] * mk;
        out[fi + CH_STRIDE] = x[fi + CH_STRIDE] * mk;
      }
    }
  }
}

// ---------------------------------------------------------------------------
extern "C" void kernel_launch(void* const* d_in, const int* in_sizes, int n_in,
                              void* d_out, int out_size, void* d_ws, size_t ws_size,
                              hipStream_t stream) {
  (void)in_sizes; (void)n_in; (void)out_size; (void)ws_size;
  int p = 0;
  auto nxt = [&]() { return (const float*)d_in[p++]; };

  const float* x = nxt();

  struct BandInP { const float *W1, *b1, *W2, *b2; } bin[10];
  for (int i = 0; i < 10; ++i) { bin[i].W1 = nxt(); bin[i].b1 = nxt(); bin[i].W2 = nxt(); bin[i].b2 = nxt(); }

  struct DirP { const float *Wih, *Whh, *bih, *bhh; };
  struct BlkP { const float *fiw, *fib, *fw, *fb; DirP d[2][2]; int nd; } blk[4];
  for (int b = 0; b < 4; ++b) {
    blk[b].nd = (b % 2 == 0) ? 2 : 1;
    blk[b].fiw = nxt(); blk[b].fib = nxt();
    for (int l = 0; l < 2; ++l)
      for (int dd = 0; dd < blk[b].nd; ++dd) {
        blk[b].d[l][dd].Wih = nxt(); blk[b].d[l][dd].Whh = nxt();
        blk[b].d[l][dd].bih = nxt(); blk[b].d[l][dd].bhh = nxt();
      }
    blk[b].fw = nxt(); blk[b].fb = nxt();
  }

  struct BandOutP { const float *W1, *b1, *W2, *b2, *W3, *b3; } bo[10];
  for (int i = 0; i < 10; ++i) {
    bo[i].W1 = nxt(); bo[i].b1 = nxt(); bo[i].W2 = nxt();
    bo[i].b2 = nxt(); bo[i].W3 = nxt(); bo[i].b3 = nxt();
  }

  // workspace carve-up (floats): z 1.28M, bufA 1.28M, l1 2.56M, l2 2.56M,
  // xp 5.12M, H1 0.512M, H2 0.512M  -> ~55 MB total
  float* ws = (float*)d_ws;
  float* z    = ws;
  float* bufA = z + 1280000;
  float* l1   = bufA + 1280000;
  float* l2   = l1 + 2560000;
  float* xp   = l2 + 2560000;
  float* H1   = xp + 5120000;
  float* H2   = H1 + 512000;
  float* out  = (float*)d_out;

  static const int offs[10] = {0, 10, 20, 40, 80, 160, 320, 640, 1280, 2560};
  static const int tw[10]   = {10, 10, 20, 40, 80, 160, 320, 640, 1280, 1538}; // = 2w

  // ---- 1) band split + input MLPs -> z (t-major, band-minor, 64 feats) ----
  for (int b = 0; b < 10; ++b)
    band_in_kernel<<<125, 128, 0, stream>>>(x, bin[b].W1, bin[b].b1, bin[b].W2, bin[b].b2,
                                            z, b, offs[b], tw[b]);

  // ---- 2) four residual RNN blocks ----
  for (int b = 0; b < 4; ++b) {
    // fc_in: z -> bufA  (20000 x 64 @ 64x64)
    gemm_kernel<64><<<dim3(1250, 1), 128, 0, stream>>>(z, 64, 0, blk[b].fiw, 64,
                                                       blk[b].fib, nullptr, nullptr, 0,
                                                       bufA, 64, 0);
    if (blk[b].nd == 2) {  // bandwise bidirectional
      for (int dd = 0; dd < 2; ++dd)
        lstm_band_kernel<64><<<125, 256, 0, stream>>>(bufA,
            blk[b].d[0][dd].Wih, blk[b].d[0][dd].Whh, blk[b].d[0][dd].bih, blk[b].d[0][dd].bhh,
            l1, 128, dd * 64, dd);
      for (int dd = 0; dd < 2; ++dd)
        lstm_band_kernel<128><<<125, 256, 0, stream>>>(l1,
            blk[b].d[1][dd].Wih, blk[b].d[1][dd].Whh, blk[b].d[1][dd].bih, blk[b].d[1][dd].bhh,
            l2, 128, dd * 64, dd);
      // fc + residual: z = l2 @ fc_w + fc_b + z
      gemm_kernel<128><<<dim3(1250, 1), 128, 0, stream>>>(l2, 128, 0, blk[b].fw, 64,
                                                          blk[b].fb, nullptr, z, 64,
                                                          z, 64, 0);
    } else {               // timewise unidirectional
      gemm_kernel<64><<<dim3(1250, 4), 128, 0, stream>>>(bufA, 64, 0, blk[b].d[0][0].Wih, 256,
                                                         blk[b].d[0][0].bih, blk[b].d[0][0].bhh,
                                                         nullptr, 0, xp, 256, 0);
      lstm_time_kernel<<<1, 256, 0, stream>>>(xp, blk[b].d[0][0].Whh, l1);
      gemm_kernel<64><<<dim3(1250, 4), 128, 0, stream>>>(l1, 64, 0, blk[b].d[1][0].Wih, 256,
                                                         blk[b].d[1][0].bih, blk[b].d[1][0].bhh,
                                                         nullptr, 0, xp, 256, 0);
      lstm_time_kernel<<<1, 256, 0, stream>>>(xp, blk[b].d[1][0].Whh, l2);
      gemm_kernel<64><<<dim3(1250, 1), 128, 0, stream>>>(l2, 64, 0, blk[b].fw, 64,
                                                         blk[b].fb, nullptr, z, 64,
                                                         z, 64, 0);
    }
  }

  // ---- 3) per-band mask heads + apply ----
  for (int b = 0; b < 10; ++b) {
    // H1 = lrelu(z[:, :, b, :] @ W1 + b1)   (2000 x 256, K=64)
    gemm_kernel<64><<<dim3(125, 4), 128, 0, stream>>>(z, 640, b * 64, bo[b].W1, 256,
                                                      bo[b].b1, nullptr, nullptr, 0,
                                                      H1, 256, 1);
    // H2 = lrelu(H1 @ W2 + b2)              (2000 x 256, K=256)
    gemm_kernel<256><<<dim3(125, 4), 128, 0, stream>>>(H1, 256, 0, bo[b].W2, 256,
                                                       bo[b].b2, nullptr, nullptr, 0,
                                                       H2, 256, 1);
    // H3 + GLU + mask*x -> out
    int halfN = tw[b];              // 2w
    int gy = (halfN + 63) / 64;
    maskout_kernel<<<dim3(125, gy), 128, 0, stream>>>(H2, bo[b].W3, 2 * halfN, bo[b].b3,
                                                      x, out, offs[b], halfN);
  }
}